// Sparsemax_90555090469645
// MI455X (gfx1250) — compile-verified
//
#include <hip/hip_runtime.h>
#include <hip/hip_bf16.h>

// Sparsemax, one row per workgroup. Memory-bound: read row once via the CDNA5
// async global->LDS data mover, run register-resident Michelot threshold
// iteration (exact, no sort), write once.

#define NCOLS 8192
#define TPB 256
#define NWAVES (TPB / 32)
#define PER_THREAD (NCOLS / TPB)        // 32 floats / thread
#define VEC_PER_THREAD (PER_THREAD / 4) // 8 float4 / thread

__device__ __forceinline__ float wave_sum(float v) {
#pragma unroll
  for (int off = 16; off > 0; off >>= 1) v += __shfl_xor(v, off, 32);
  return v;
}

__global__ __launch_bounds__(TPB) void sparsemax_kernel(const float* __restrict__ x,
                                                        float* __restrict__ out) {
  __shared__ float s_row[NCOLS];          // 32 KB row staging
  __shared__ float s_red[2][NWAVES];      // (sum, count) partials per wave

  const int tid  = threadIdx.x;
  const int lane = tid & 31;
  const int wave = tid >> 5;
  const int row  = blockIdx.x;

  const float* rowp = x + (size_t)row * NCOLS;   // uniform -> SGPR pair

  // ---- CDNA5 async copy: global -> LDS, GVS mode, tracked by ASYNCcnt ----
  // dsaddr = lds_vgpr + inst_offset ; global = saddr + vaddr + inst_offset
  // Flat-address low 32 bits of a __shared__ pointer == LDS byte offset.
  unsigned lds_addr = (unsigned)(uintptr_t)(&s_row[0]) + (unsigned)tid * 16u;
  unsigned goff     = (unsigned)tid * 16u;

#define ASYNC_ROW_CP(OFF)                                                      \
  asm volatile("global_load_async_to_lds_b128 %0, %1, %2 offset:" #OFF         \
               :: "v"(lds_addr), "v"(goff), "s"(rowp) : "memory")
  ASYNC_ROW_CP(0);
  ASYNC_ROW_CP(4096);
  ASYNC_ROW_CP(8192);
  ASYNC_ROW_CP(12288);
  ASYNC_ROW_CP(16384);
  ASYNC_ROW_CP(20480);
  ASYNC_ROW_CP(24576);
  ASYNC_ROW_CP(28672);
#undef ASYNC_ROW_CP

  asm volatile("s_wait_asynccnt 0" ::: "memory");
  __syncthreads();

  // ---- row -> registers, coalesced float4 layout (chunk j, lane-contiguous) ----
  float4 v[VEC_PER_THREAD];
  const float4* s4 = (const float4*)s_row;
#pragma unroll
  for (int j = 0; j < VEC_PER_THREAD; ++j) v[j] = s4[j * TPB + tid];

  // ---- block-wide (sum, count) reduction helper ----
  float S, K;
#define BLOCK_REDUCE2(a_in, b_in)                                              \
  do {                                                                         \
    float _a = wave_sum(a_in), _b = wave_sum(b_in);                            \
    if (lane == 0) { s_red[0][wave] = _a; s_red[1][wave] = _b; }               \
    __syncthreads();                                                           \
    S = 0.0f; K = 0.0f;                                                        \
    _Pragma("unroll")                                                          \
    for (int w = 0; w < NWAVES; ++w) { S += s_red[0][w]; K += s_red[1][w]; }   \
    __syncthreads();                                                           \
  } while (0)

  // total row sum -> initial threshold tau0 = (sum - 1)/N
  {
    float loc = 0.0f;
#pragma unroll
    for (int j = 0; j < VEC_PER_THREAD; ++j)
      loc += (v[j].x + v[j].y) + (v[j].z + v[j].w);
    BLOCK_REDUCE2(loc, 0.0f);
  }
  float tau = (S - 1.0f) * (1.0f / (float)NCOLS);

  // ---- Michelot fixed-point iteration: tau' = (sum{x>tau} - 1)/count{x>tau}.
  // tau increases monotonically; fixed point == exact sparsemax threshold.
  // All lanes take identical (uniform) branches; capped for determinism.
  for (int it = 0; it < 64; ++it) {
    float s = 0.0f, k = 0.0f;
#pragma unroll
    for (int j = 0; j < VEC_PER_THREAD; ++j) {
      float m;
      m = (v[j].x > tau) ? 1.0f : 0.0f; s = fmaf(v[j].x, m, s); k += m;
      m = (v[j].y > tau) ? 1.0f : 0.0f; s = fmaf(v[j].y, m, s); k += m;
      m = (v[j].z > tau) ? 1.0f : 0.0f; s = fmaf(v[j].z, m, s); k += m;
      m = (v[j].w > tau) ? 1.0f : 0.0f; s = fmaf(v[j].w, m, s); k += m;
    }
    BLOCK_REDUCE2(s, k);
    float nt = (S - 1.0f) / K;   // K >= 1 always (max element stays above tau)
    if (nt == tau) break;        // support set stabilized -> exact tau
    tau = nt;
  }
#undef BLOCK_REDUCE2

  // ---- out = relu(x - tau), coalesced float4 stores ----
  float4* o4 = (float4*)(out + (size_t)row * NCOLS);
#pragma unroll
  for (int j = 0; j < VEC_PER_THREAD; ++j) {
    float4 r;
    r.x = fmaxf(v[j].x - tau, 0.0f);
    r.y = fmaxf(v[j].y - tau, 0.0f);
    r.z = fmaxf(v[j].z - tau, 0.0f);
    r.w = fmaxf(v[j].w - tau, 0.0f);
    o4[j * TPB + tid] = r;
  }
}

extern "C" void kernel_launch(void* const* d_in, const int* in_sizes, int n_in,
                              void* d_out, int out_size, void* d_ws, size_t ws_size,
                              hipStream_t stream) {
  const float* x = (const float*)d_in[0];
  float* out = (float*)d_out;
  const int rows = in_sizes[0] / NCOLS;  // 4096
  sparsemax_kernel<<<dim3(rows), dim3(TPB), 0, stream>>>(x, out);
}